// GCN_18665927868953
// MI455X (gfx1250) — compile-verified
//
#include <hip/hip_runtime.h>

#define HF 32   // hidden features
#define CLS 3   // classes

typedef float v2f __attribute__((ext_vector_type(2)));
typedef float v8f __attribute__((ext_vector_type(8)));

// ---------------- utility ----------------
__global__ void zero_f32(float* p, int n) {
    int i = blockIdx.x * blockDim.x + threadIdx.x;
    if (i < n) p[i] = 0.0f;
}

// deg[dst] += 1 per edge (self-loop folded in later as +1)
__global__ void deg_count(const int* __restrict__ dst, float* __restrict__ deg, int E) {
    int e = blockIdx.x * blockDim.x + threadIdx.x;
    if (e < E) atomicAdd(&deg[dst[e]], 1.0f);
}

// dinv[i] = rsqrt(deg[i] + 1)   (in place)
__global__ void make_dinv(float* __restrict__ d, int n) {
    int i = blockIdx.x * blockDim.x + threadIdx.x;
    if (i < n) d[i] = rsqrtf(d[i] + 1.0f);
}

// scalar aggregation for layer 1: sagg[dst] += dinv[s]*dinv[d]*x[s]
__global__ void agg_scalar(const int* __restrict__ ei, const float* __restrict__ x,
                           const float* __restrict__ dinv, float* __restrict__ sagg, int E) {
    int e = blockIdx.x * blockDim.x + threadIdx.x;
    if (e >= E) return;
    int s = ei[e], d = ei[E + e];
    atomicAdd(&sagg[d], dinv[s] * dinv[d] * x[s]);
}

// out1[i][j] = relu((sagg[i] + dinv[i]^2 * x[i]) * W1[j] + b1[j])
__global__ void expand1(const float* __restrict__ x, const float* __restrict__ dinv,
                        const float* __restrict__ sagg, const float* __restrict__ W1,
                        const float* __restrict__ b1, float* __restrict__ out, int n) {
    int t = blockIdx.x * blockDim.x + threadIdx.x;
    if (t >= n * HF) return;
    int i = t >> 5, j = t & 31;
    float di = dinv[i];
    float s = sagg[i] + di * di * x[i];
    float v = fmaf(s, W1[j], b1[j]);
    out[t] = v > 0.0f ? v : 0.0f;
}

// 32-feature edge aggregation: out[dst] += dinv[s]*dinv[d] * in[src]
// 8 threads per edge, float4 gather, 4 atomic f32 scatters each.
__global__ void agg_feat(const int* __restrict__ ei, const float* __restrict__ dinv,
                         const float* __restrict__ in, float* out, int E) {
    long long t = (long long)blockIdx.x * blockDim.x + threadIdx.x;
    int e = (int)(t >> 3);
    if (e >= E) return;
    int sub = (int)(t & 7);
    int s = ei[e], d = ei[E + e];
    float w = dinv[s] * dinv[d];
    const float4 v = *(const float4*)(in + s * HF + sub * 4);
    float* o = out + d * HF + sub * 4;
    atomicAdd(o + 0, w * v.x);
    atomicAdd(o + 1, w * v.y);
    atomicAdd(o + 2, w * v.z);
    atomicAdd(o + 3, w * v.w);
}

// Fused GEMM: Out[row] = act( (Aagg[row] + dinv[row]^2 * Aself[row]) @ Wm + bias )
// One wave per 16-row tile; V_WMMA_F32_16X16X4_F32, K=32 in 8 steps, 2 column halves.
// A 16x4 f32 layout: element [m][k] -> lane = m + 16*(k>>1), reg = k&1.
// B  4x16 f32 layout: element [k][n] -> lane = n + 16*(k>>1), reg = k&1.
// C/D 16x16 f32: reg r, lane -> row = r + 8*(lane>>4), col = lane&15.
// Out may alias Aself (row-exclusive per wave; all reads precede stores via data deps).
__global__ void gemm_h32_wmma(const float* __restrict__ Aagg, const float* Aself,
                              const float* __restrict__ dinv, const float* __restrict__ Wm,
                              const float* __restrict__ bias, float* Out,
                              int nrows, int relu) {
    int gtid = blockIdx.x * blockDim.x + threadIdx.x;
    int tile = gtid >> 5;               // one wave32 per tile
    int row0 = tile * 16;
    if (row0 >= nrows) return;          // whole wave exits together (EXEC stays all-1)
    int lane = threadIdx.x & 31;
    int half = lane >> 4;
    int mn   = lane & 15;

    int arow = row0 + mn;
    float di = dinv[arow];
    float d2 = di * di;
    const float* ap = Aagg  + arow * HF;
    const float* sp = Aself + arow * HF;

    v8f c0 = {};
    v8f c1 = {};
#pragma unroll
    for (int kk = 0; kk < HF; kk += 4) {
        int k0 = kk + 2 * half;
        v2f a, b0, b1;
        a[0] = fmaf(d2, sp[k0    ], ap[k0    ]);
        a[1] = fmaf(d2, sp[k0 + 1], ap[k0 + 1]);
        b0[0] = Wm[k0 * HF + mn];
        b0[1] = Wm[(k0 + 1) * HF + mn];
        b1[0] = Wm[k0 * HF + 16 + mn];
        b1[1] = Wm[(k0 + 1) * HF + 16 + mn];
        c0 = __builtin_amdgcn_wmma_f32_16x16x4_f32(false, a, false, b0, (short)0, c0, false, false);
        c1 = __builtin_amdgcn_wmma_f32_16x16x4_f32(false, a, false, b1, (short)0, c1, false, false);
    }
#pragma unroll
    for (int r = 0; r < 8; r++) {
        int orow = row0 + r + 8 * half;
        float v0 = c0[r] + bias[mn];
        float v1 = c1[r] + bias[16 + mn];
        if (relu) { v0 = fmaxf(v0, 0.0f); v1 = fmaxf(v1, 0.0f); }
        Out[orow * HF + mn]      = v0;
        Out[orow * HF + 16 + mn] = v1;
    }
}

// per-graph pooling: psum[batch[i]] += h[i]; pcnt[batch[i]] += 1
__global__ void pool_sum(const float* __restrict__ h, const int* __restrict__ batch,
                         float* psum, float* pcnt, int n) {
    long long t = (long long)blockIdx.x * blockDim.x + threadIdx.x;
    int i = (int)(t >> 3);
    if (i >= n) return;
    int sub = (int)(t & 7);
    int g = batch[i];
    const float4 v = *(const float4*)(h + i * HF + sub * 4);
    float* o = psum + g * HF + sub * 4;
    atomicAdd(o + 0, v.x);
    atomicAdd(o + 1, v.y);
    atomicAdd(o + 2, v.z);
    atomicAdd(o + 3, v.w);
    if (sub == 0) atomicAdd(&pcnt[g], 1.0f);
}

// out[g][c] = mean(psum[g]) @ Wl + bl
__global__ void final_linear(const float* __restrict__ psum, const float* __restrict__ pcnt,
                             const float* __restrict__ Wl, const float* __restrict__ bl,
                             float* __restrict__ out, int G) {
    int g = blockIdx.x * blockDim.x + threadIdx.x;
    if (g >= G) return;
    float inv = 1.0f / fmaxf(pcnt[g], 1.0f);
#pragma unroll
    for (int c = 0; c < CLS; c++) {
        float acc = bl[c];
#pragma unroll
        for (int j = 0; j < HF; j++)
            acc = fmaf(psum[g * HF + j] * inv, Wl[j * CLS + c], acc);
        out[g * CLS + c] = acc;
    }
}

extern "C" void kernel_launch(void* const* d_in, const int* in_sizes, int n_in,
                              void* d_out, int out_size, void* d_ws, size_t ws_size,
                              hipStream_t stream) {
    const float* x     = (const float*)d_in[0];
    const int*   ei    = (const int*)d_in[1];   // [2,E]
    const int*   batch = (const int*)d_in[2];   // [N]
    const float* W1 = (const float*)d_in[3];
    const float* b1 = (const float*)d_in[4];
    const float* W2 = (const float*)d_in[5];
    const float* b2 = (const float*)d_in[6];
    const float* W3 = (const float*)d_in[7];
    const float* b3 = (const float*)d_in[8];
    const float* Wl = (const float*)d_in[9];
    const float* bl = (const float*)d_in[10];

    const int N = in_sizes[0];
    const int E = in_sizes[1] / 2;
    const int G = out_size / CLS;

    float* ws   = (float*)d_ws;
    float* dinv = ws;                              // N   (holds deg first)
    float* sagg = dinv + N;                        // N
    float* bufA = sagg + N;                        // N*HF
    float* bufB = bufA + (size_t)N * HF;           // N*HF
    float* psum = bufB + (size_t)N * HF;           // G*HF
    float* pcnt = psum + (size_t)G * HF;           // G
    float* out  = (float*)d_out;

    const int T = 256;
    auto nb = [](long long n, int t) { return (int)((n + t - 1) / t); };

    // norm
    zero_f32 <<<nb(2LL * N, T), T, 0, stream>>>(dinv, 2 * N);            // deg + sagg
    deg_count<<<nb(E, T), T, 0, stream>>>(ei + E, dinv, E);
    make_dinv<<<nb(N, T), T, 0, stream>>>(dinv, N);

    // layer 1 (scalar aggregate, then expand by W1)
    agg_scalar<<<nb(E, T), T, 0, stream>>>(ei, x, dinv, sagg, E);
    expand1  <<<nb((long long)N * HF, T), T, 0, stream>>>(x, dinv, sagg, W1, b1, bufA, N);

    const int tiles = (N + 15) / 16;
    const int gemm_blocks = nb((long long)tiles * 32, T);

    // layer 2
    zero_f32 <<<nb((long long)N * HF, T), T, 0, stream>>>(bufB, N * HF);
    agg_feat <<<nb(8LL * E, T), T, 0, stream>>>(ei, dinv, bufA, bufB, E);
    gemm_h32_wmma<<<gemm_blocks, T, 0, stream>>>(bufB, bufA, dinv, W2, b2, bufA, N, 1);

    // layer 3
    zero_f32 <<<nb((long long)N * HF, T), T, 0, stream>>>(bufB, N * HF);
    agg_feat <<<nb(8LL * E, T), T, 0, stream>>>(ei, dinv, bufA, bufB, E);
    gemm_h32_wmma<<<gemm_blocks, T, 0, stream>>>(bufB, bufA, dinv, W3, b3, bufA, N, 0);

    // mean pool + classifier
    zero_f32 <<<nb((long long)G * (HF + 1), T), T, 0, stream>>>(psum, G * (HF + 1));
    pool_sum <<<nb(8LL * N, T), T, 0, stream>>>(bufA, batch, psum, pcnt, N);
    final_linear<<<nb(G, T), T, 0, stream>>>(psum, pcnt, Wl, bl, out, G);
}